// SmallLiquidStateMachineSNN_27616639713863
// MI455X (gfx1250) — compile-verified
//
#include <hip/hip_runtime.h>
#include <stdint.h>

typedef __attribute__((ext_vector_type(2))) float v2f;
typedef __attribute__((ext_vector_type(8))) float v8f;

#define T_STEPS 256
#define BATCH   64
#define DIN     512
#define RDIM    2048
#define DOUT    128
#define N_W     (RDIM * RDIM)      // 4194304
#define KEEP    838860             // int(N_W * (1-0.8)) as in reference
#define BETA    0.9f
#define VTHRESH 1.0f

#define KC      32                 // K-strip depth staged in LDS
#define PAD     40                 // LDS row pitch (floats): 2*PAD mod 64 = 16 -> conflict-free halves, 16B aligned
#define NSTRIPS ((DIN + RDIM) / KC)   // 80 strips: 16 of w_in, 64 of w_eff

// ---------------- radix-select (k-th largest of |w_rec|) ----------------

__global__ void init_misc(unsigned* sel, unsigned long long* spike_cnt,
                          unsigned long long* active_cnt) {
    sel[0] = 0u;          // prefix (high bytes fixed so far)
    sel[1] = KEEP;        // remaining rank (1-based, from the top)
    *spike_cnt  = 0ull;
    *active_cnt = 0ull;
}

__global__ void init_state(float* V, float* s0, float* s1) {
    int i = blockIdx.x * blockDim.x + threadIdx.x;   // 512*256 == 131072 exactly
    V[i]  = 0.0f;
    s0[i] = 0.0f;
    s1[i] = 0.0f;
}

__global__ void zero_hist(unsigned* hist) { hist[threadIdx.x] = 0u; }

__global__ __launch_bounds__(256) void hist_pass(const float* __restrict__ w,
                                                 unsigned* __restrict__ hist,
                                                 const unsigned* __restrict__ sel,
                                                 int pass) {
    __shared__ unsigned sh[256];
    sh[threadIdx.x] = 0u;
    __syncthreads();
    const unsigned hiMask = (pass == 3) ? 0u : (0xFFFFFFFFu << ((pass + 1) * 8));
    const unsigned prefix = sel[0];
    const int shift = pass * 8;
    int stride = gridDim.x * blockDim.x;
    for (int i = blockIdx.x * blockDim.x + threadIdx.x; i < N_W; i += stride) {
        unsigned u = __float_as_uint(fabsf(w[i]));
        if ((u & hiMask) == (prefix & hiMask))
            atomicAdd(&sh[(u >> shift) & 255u], 1u);
    }
    __syncthreads();
    if (sh[threadIdx.x]) atomicAdd(&hist[threadIdx.x], sh[threadIdx.x]);
}

__global__ void select_pass(const unsigned* __restrict__ hist, unsigned* sel, int pass) {
    unsigned K = sel[1];
    unsigned cum = 0;
    for (int b = 255; b >= 0; --b) {
        unsigned c = hist[b];
        if (cum + c >= K) {
            sel[0] |= ((unsigned)b) << (pass * 8);
            sel[1] = K - cum;
            return;
        }
        cum += c;
    }
}

__global__ __launch_bounds__(256) void build_mask(const float* __restrict__ w,
                                                  float* __restrict__ w_eff,
                                                  const unsigned* __restrict__ sel,
                                                  unsigned long long* active_cnt) {
    __shared__ unsigned sh[256];
    const float thr = __uint_as_float(sel[0]);
    unsigned cnt = 0;
    int stride = gridDim.x * blockDim.x;
    for (int i = blockIdx.x * blockDim.x + threadIdx.x; i < N_W; i += stride) {
        float wv = w[i];
        bool keep = fabsf(wv) >= thr;
        w_eff[i] = keep ? wv : 0.0f;
        cnt += keep ? 1u : 0u;
    }
    sh[threadIdx.x] = cnt;
    __syncthreads();
    for (int off = 128; off > 0; off >>= 1) {
        if (threadIdx.x < off) sh[threadIdx.x] += sh[threadIdx.x + off];
        __syncthreads();
    }
    if (threadIdx.x == 0) atomicAdd(active_cnt, (unsigned long long)sh[0]);
}

// ---------------- per-timestep LIF step: fused input + recurrent WMMA ----------------
// Block = 256 threads = 8 wave32 -> 64x32 output strip (4 m-tiles x 2 n-tiles).
// Virtual B of 2560 rows (w_in then w_eff) staged in LDS in 80 strips of 32x32,
// double-buffered with GLOBAL_LOAD_ASYNC_TO_LDS_B128 + s_wait_asynccnt.

__global__ __launch_bounds__(256) void snn_step(const float* __restrict__ x_t,
                                                const float* __restrict__ w_in,
                                                const float* __restrict__ b_in,
                                                const float* __restrict__ w_eff,
                                                const float* __restrict__ s_in,
                                                float* __restrict__ s_out,
                                                float* __restrict__ V,
                                                unsigned long long* __restrict__ spike_cnt) {
    __shared__ float shb[2][KC * PAD];     // 2 x 5120 B double buffer
    __shared__ unsigned shred[256];

    const int lane = threadIdx.x & 31;
    const int wave = threadIdx.x >> 5;
    const int m0 = (wave & 3) << 4;        // batch tile origin
    const int nl = ((wave >> 2) << 4) + (lane & 15);   // local col 0..31
    const int c0 = blockIdx.x * 32;        // global col strip origin
    const int l16 = lane & 15;
    const int hi  = lane >> 4;             // 0: K=0,1   1: K=2,3

    // staging assignment: thread -> one 16B chunk of the 32x32 strip
    const int strow = threadIdx.x >> 3;    // 0..31
    const int stcol = (threadIdx.x & 7) * 4;
    const unsigned lds0 = (unsigned)(uintptr_t)&shb[0][strow * PAD + stcol];
    const unsigned lds1 = (unsigned)(uintptr_t)&shb[1][strow * PAD + stcol];

    const float* xrow = x_t  + (size_t)(m0 + l16) * DIN;
    const float* srow = s_in + (size_t)(m0 + l16) * RDIM;

    auto issue = [&](int strip, int buf) {
        const float* gB = (strip < DIN / KC)
                            ? (w_in  + (size_t)strip * KC * RDIM)
                            : (w_eff + (size_t)(strip - DIN / KC) * KC * RDIM);
        const float* ga = gB + (size_t)strow * RDIM + c0 + stcol;
        unsigned long long gaddr = (unsigned long long)(uintptr_t)ga;
        unsigned lds = buf ? lds1 : lds0;
        asm volatile("global_load_async_to_lds_b128 %0, %1, off"
                     :: "v"(lds), "v"(gaddr) : "memory");
    };

    v8f acc = {};
    issue(0, 0);
    for (int strip = 0; strip < NSTRIPS; ++strip) {
        const int cur = strip & 1;
        if (strip + 1 < NSTRIPS) {
            issue(strip + 1, cur ^ 1);                 // prefetch next strip
            asm volatile("s_wait_asynccnt 1" ::: "memory");  // strip `strip` landed
        } else {
            asm volatile("s_wait_asynccnt 0" ::: "memory");
        }
        __syncthreads();

        const float* aptr = (strip < DIN / KC) ? (xrow + strip * KC)
                                               : (srow + (strip - DIN / KC) * KC);
        const float* bb = &shb[cur][0];
        #pragma unroll
        for (int kk = 0; kk < KC; kk += 4) {
            const int kb = kk + 2 * hi;
            v2f a = *(const v2f*)(aptr + kb);
            v2f b = { bb[kb * PAD + nl], bb[(kb + 1) * PAD + nl] };
            acc = __builtin_amdgcn_wmma_f32_16x16x4_f32(false, a, false, b,
                                                        (short)0, acc, false, false);
        }
        __syncthreads();                               // protect buffer reuse
    }

    const int n = c0 + nl;
    const float bias = b_in[n];
    unsigned spikes = 0;
    #pragma unroll
    for (int r = 0; r < 8; ++r) {
        const int m = m0 + r + 8 * hi;                 // C/D layout: VGPR r -> M=r / r+8
        const int idx = m * RDIM + n;
        float v = BETA * V[idx] + acc[r] + bias;       // leaky integrate
        float s = (v > VTHRESH) ? 1.0f : 0.0f;         // heaviside spike
        v -= s * VTHRESH;                              // soft reset
        V[idx]     = v;
        s_out[idx] = s;
        spikes += (unsigned)s;
    }

    shred[threadIdx.x] = spikes;
    __syncthreads();
    for (int off = 128; off > 0; off >>= 1) {
        if (threadIdx.x < off) shred[threadIdx.x] += shred[threadIdx.x + off];
        __syncthreads();
    }
    if (threadIdx.x == 0) atomicAdd(spike_cnt, (unsigned long long)shred[0]);
}

// ---------------- head: logits_t[64x128] = s @ w_head + b_head ----------------
// 4x8 = 32 tiles -> 4 blocks * 8 waves

__global__ __launch_bounds__(256) void snn_head(const float* __restrict__ s,
                                                const float* __restrict__ w_head,
                                                const float* __restrict__ b_head,
                                                float* __restrict__ logits_t) {
    const int lane = threadIdx.x & 31;
    const int wave = threadIdx.x >> 5;
    const int tile = blockIdx.x * 8 + wave;      // 0..31
    const int m0 = (tile & 3) << 4;
    const int n0 = (tile >> 2) << 4;             // 0..112
    const int l16 = lane & 15;
    const int hi  = lane >> 4;

    v8f acc = {};
    const float* srow = s + (size_t)(m0 + l16) * RDIM;
    #pragma unroll 2
    for (int k = 0; k < RDIM; k += 4) {
        const int kb = k + 2 * hi;
        v2f a = *(const v2f*)(srow + kb);
        v2f b = { w_head[(size_t)kb * DOUT + n0 + l16],
                  w_head[(size_t)(kb + 1) * DOUT + n0 + l16] };
        acc = __builtin_amdgcn_wmma_f32_16x16x4_f32(false, a, false, b,
                                                    (short)0, acc, false, false);
    }

    const float bias = b_head[n0 + l16];
    const int n = n0 + l16;
    #pragma unroll
    for (int r = 0; r < 8; ++r) {
        const int m = m0 + r + 8 * hi;
        logits_t[m * DOUT + n] = acc[r] + bias;
    }
}

// ---------------- finalize: readout mean, spike_rate, active_ratio ----------------

__global__ __launch_bounds__(256) void finalize(float* __restrict__ out,
                                                const float* __restrict__ logits_base,
                                                const unsigned long long* __restrict__ spike_cnt,
                                                const unsigned long long* __restrict__ active_cnt) {
    const int e = blockIdx.x * blockDim.x + threadIdx.x;   // 32*256 = 8192
    float sum = 0.0f;
    for (int t = 0; t < T_STEPS; ++t)
        sum += logits_base[(size_t)t * (BATCH * DOUT) + e];
    out[e] = sum * (1.0f / T_STEPS);
    if (e == 0) {
        const size_t base = (size_t)BATCH * DOUT + (size_t)T_STEPS * BATCH * DOUT;
        out[base]     = (float)((double)*spike_cnt / (double)((size_t)T_STEPS * BATCH * RDIM));
        out[base + 1] = (float)((double)*active_cnt / (double)N_W);
    }
}

// ---------------- host launcher ----------------

extern "C" void kernel_launch(void* const* d_in, const int* in_sizes, int n_in,
                              void* d_out, int out_size, void* d_ws, size_t ws_size,
                              hipStream_t stream) {
    const float* x_seq  = (const float*)d_in[0];   // [T,B,DIN]
    const float* w_in   = (const float*)d_in[1];   // [DIN,R]
    const float* b_in   = (const float*)d_in[2];   // [R]
    const float* w_rec  = (const float*)d_in[3];   // [R,R]
    const float* w_head = (const float*)d_in[4];   // [R,DOUT]
    const float* b_head = (const float*)d_in[5];   // [DOUT]
    float* out = (float*)d_out;

    char* ws = (char*)d_ws;
    float* w_eff = (float*)ws;                             // 16 MB
    float* V     = (float*)(ws + (size_t)N_W * 4);         // 512 KB
    float* s0    = V + BATCH * RDIM;
    float* s1    = s0 + BATCH * RDIM;
    unsigned* hist = (unsigned*)(s1 + BATCH * RDIM);       // 256 u32
    unsigned* sel  = hist + 256;                           // prefix, remaining-K
    unsigned long long* spike_cnt  = (unsigned long long*)(sel + 4);
    unsigned long long* active_cnt = spike_cnt + 1;

    init_misc<<<1, 1, 0, stream>>>(sel, spike_cnt, active_cnt);
    init_state<<<512, 256, 0, stream>>>(V, s0, s1);

    for (int pass = 3; pass >= 0; --pass) {
        zero_hist<<<1, 256, 0, stream>>>(hist);
        hist_pass<<<2048, 256, 0, stream>>>(w_rec, hist, sel, pass);
        select_pass<<<1, 1, 0, stream>>>(hist, sel, pass);
    }
    build_mask<<<2048, 256, 0, stream>>>(w_rec, w_eff, sel, active_cnt);

    float* logits_base = out + BATCH * DOUT;
    for (int t = 0; t < T_STEPS; ++t) {
        const float* s_in = (t & 1) ? s1 : s0;
        float* s_out_buf  = (t & 1) ? s0 : s1;
        snn_step<<<64, 256, 0, stream>>>(x_seq + (size_t)t * BATCH * DIN,
                                         w_in, b_in, w_eff, s_in, s_out_buf,
                                         V, spike_cnt);
        snn_head<<<4, 256, 0, stream>>>(s_out_buf, w_head, b_head,
                                        logits_base + (size_t)t * BATCH * DOUT);
    }
    finalize<<<32, 256, 0, stream>>>(out, logits_base, spike_cnt, active_cnt);
}